// SPCS_NoSPP_10299331575814
// MI455X (gfx1250) — compile-verified
//
#include <hip/hip_runtime.h>
#include <hip/hip_bf16.h>
#include <math.h>

typedef __attribute__((ext_vector_type(2))) float v2f;
typedef __attribute__((ext_vector_type(8))) float v8f;

#define N_   16
#define C_   256
#define H_   56
#define W_   56
#define MIP  8
#define L_   112
#define BN_EPS 1e-5f

// -------------------------------------------------------------------------
// Kernel 1: per-(n,c) plane pooling. x_h = mean over W, x_w = mean over H.
// pooled layout: (N, C, 112) = [x_h(56) | x_w(56)] matching the reference
// concat. One 64-thread block (2 waves) per plane; lane t owns column t.
// -------------------------------------------------------------------------
__global__ __launch_bounds__(64) void pool_kernel(const float* __restrict__ x,
                                                  float* __restrict__ pooled) {
    const int nc   = blockIdx.x;           // n*C + c
    const int t    = threadIdx.x;          // 0..63
    const int wave = t >> 5;
    const int lane = t & 31;
    const float* p = x + (size_t)nc * (H_ * W_);

    __shared__ float rowpart[2][H_];

    const bool colvalid = (t < W_);
    const int  tcol     = colvalid ? t : 0;
    float colsum = 0.f;

    for (int h = 0; h < H_; ++h) {
        float v = p[h * W_ + tcol];        // coalesced row load
        v = colvalid ? v : 0.f;
        colsum += v;
        // wave32 all-reduce of this row's 32-lane slice
        float r = v;
        #pragma unroll
        for (int m = 16; m >= 1; m >>= 1) r += __shfl_xor(r, m, 32);
        if (lane == 0) rowpart[wave][h] = r;
    }
    __syncthreads();

    if (t < H_) {
        float rowsum = rowpart[0][t] + rowpart[1][t];
        pooled[(size_t)nc * L_ + t] = rowsum * (1.f / W_);          // x_h
    }
    if (t < W_) {
        pooled[(size_t)nc * L_ + H_ + t] = colsum * (1.f / H_);     // x_w
    }
}

// -------------------------------------------------------------------------
// Kernel 2: y = h_swish(BN(conv1_w @ pooled + b)).
// Per block (one wave): one (n, 16-wide l-tile). D(16x16) = A(16x4)B(4x16)
// accumulated over K=256 with V_WMMA_F32_16X16X4_F32. A rows 8..15 zeroed.
// -------------------------------------------------------------------------
__global__ __launch_bounds__(32) void conv1_kernel(
    const float* __restrict__ pooled,
    const float* __restrict__ w1, const float* __restrict__ b1,
    const float* __restrict__ gamma, const float* __restrict__ beta,
    const float* __restrict__ mean, const float* __restrict__ var,
    float* __restrict__ ybuf) {
    const int nt    = blockIdx.x % 7;      // l-tile
    const int n     = blockIdx.x / 7;
    const int lane  = threadIdx.x;
    const int idx16 = lane & 15;
    const int khalf = lane >> 4;           // 0 or 1 -> K pair {0,1} or {2,3}
    const int rowA  = idx16;               // A row m; valid only if < 8
    const int colB  = nt * 16 + idx16;     // output column l in [0,112)
    const float* pb = pooled + (size_t)n * C_ * L_;

    v8f acc = {};
    #pragma unroll 4
    for (int k0 = 0; k0 < C_; k0 += 4) {
        const int ka = k0 + 2 * khalf;
        v2f a, b;
        const float a0 = w1[(rowA & 7) * C_ + ka];       // clamped addr, no OOB
        const float a1 = w1[(rowA & 7) * C_ + ka + 1];
        a.x = (rowA < MIP) ? a0 : 0.f;                   // zero-pad rows 8..15
        a.y = (rowA < MIP) ? a1 : 0.f;
        b.x = pb[(size_t)ka * L_ + colB];
        b.y = pb[(size_t)(ka + 1) * L_ + colB];
        acc = __builtin_amdgcn_wmma_f32_16x16x4_f32(
            false, a, false, b, (short)0, acc, false, false);
    }

    // Rows m=0..7 live in VGPRs r=0..7 of lanes 0..15 (N = lane).
    if (lane < 16) {
        #pragma unroll
        for (int r = 0; r < MIP; ++r) {
            const int m = r;
            const float inv = gamma[m] * rsqrtf(var[m] + BN_EPS);
            float v = acc[r] + b1[m];
            v = (v - mean[m]) * inv + beta[m];
            const float hs = v * fminf(fmaxf(v + 3.f, 0.f), 6.f) * (1.f / 6.f);
            ybuf[((size_t)n * MIP + m) * L_ + nt * 16 + lane] = hs;
        }
    }
}

// -------------------------------------------------------------------------
// Kernel 3: a_{h,w} = sigmoid(conv_{h,w}_w @ y_{h,w} + b).
// Per block (one wave): one (branch, n, 16-row m-tile, 16-col tile).
// K = 8 -> two V_WMMA_F32_16X16X4_F32. Cols >= 56 masked.
// -------------------------------------------------------------------------
__global__ __launch_bounds__(32) void attn_kernel(
    const float* __restrict__ ybuf,
    const float* __restrict__ wh, const float* __restrict__ bh,
    const float* __restrict__ ww, const float* __restrict__ bw,
    float* __restrict__ a_h, float* __restrict__ a_w) {
    const int mt     = blockIdx.x;         // 0..15  (256 rows / 16)
    const int nt     = blockIdx.y;         // 0..3   (56 cols -> 4 tiles)
    const int branch = blockIdx.z & 1;     // 0 = h, 1 = w
    const int n      = blockIdx.z >> 1;
    const int lane   = threadIdx.x;
    const int idx16  = lane & 15;
    const int khalf  = lane >> 4;

    const float* wmat = branch ? ww : wh;
    const float* bias = branch ? bw : bh;
    float*       aout = branch ? a_w : a_h;

    const int mA   = mt * 16 + idx16;              // A row (output channel)
    const int col  = nt * 16 + idx16;              // valid if < 56
    const int colc = (col < H_) ? col : (H_ - 1);  // clamped address
    const float* yb = ybuf + ((size_t)n * MIP) * L_ + branch * H_;

    v8f acc = {};
    #pragma unroll
    for (int k0 = 0; k0 < MIP; k0 += 4) {
        const int ka = k0 + 2 * khalf;
        v2f a, b;
        a.x = wmat[(size_t)mA * MIP + ka];
        a.y = wmat[(size_t)mA * MIP + ka + 1];
        const float b0 = yb[(size_t)ka * L_ + colc];
        const float b1v = yb[(size_t)(ka + 1) * L_ + colc];
        b.x = (col < H_) ? b0 : 0.f;
        b.y = (col < H_) ? b1v : 0.f;
        acc = __builtin_amdgcn_wmma_f32_16x16x4_f32(
            false, a, false, b, (short)0, acc, false, false);
    }

    if (col < H_) {
        #pragma unroll
        for (int r = 0; r < 8; ++r) {
            const int m = mt * 16 + r + 8 * khalf;   // C layout: M = r + 8*(lane>=16)
            const float v = acc[r] + bias[m];
            const float s = 1.f / (1.f + __expf(-v));
            aout[((size_t)n * C_ + m) * H_ + col] = s;
        }
    }
}

// -------------------------------------------------------------------------
// Kernel 4: out = x * (a_h*a_w + 1), float4-vectorized (the BW-bound pass).
// -------------------------------------------------------------------------
__global__ __launch_bounds__(256) void apply_kernel(
    const float* __restrict__ x, const float* __restrict__ a_h,
    const float* __restrict__ a_w, float* __restrict__ out) {
    const size_t i4 = (size_t)blockIdx.x * 256 + threadIdx.x;  // float4 index
    const int W4 = W_ / 4;                                     // 14
    const int w4  = (int)(i4 % W4);
    size_t rest   = i4 / W4;
    const int h   = (int)(rest % H_); rest /= H_;
    const int c   = (int)(rest % C_);
    const int n   = (int)(rest / C_);

    const float4 xv = ((const float4*)x)[i4];
    const size_t ncb = (size_t)n * C_ + c;
    const float  ah  = a_h[ncb * H_ + h];
    const float4 aw  = ((const float4*)(a_w + ncb * W_))[w4];

    float4 o;
    o.x = xv.x * (ah * aw.x + 1.f);
    o.y = xv.y * (ah * aw.y + 1.f);
    o.z = xv.z * (ah * aw.z + 1.f);
    o.w = xv.w * (ah * aw.w + 1.f);
    ((float4*)out)[i4] = o;
}

extern "C" void kernel_launch(void* const* d_in, const int* in_sizes, int n_in,
                              void* d_out, int out_size, void* d_ws, size_t ws_size,
                              hipStream_t stream) {
    const float* x    = (const float*)d_in[0];
    const float* w1   = (const float*)d_in[1];
    const float* b1   = (const float*)d_in[2];
    const float* g1   = (const float*)d_in[3];
    const float* be1  = (const float*)d_in[4];
    const float* mu1  = (const float*)d_in[5];
    const float* var1 = (const float*)d_in[6];
    const float* whw  = (const float*)d_in[7];
    const float* bhb  = (const float*)d_in[8];
    const float* www  = (const float*)d_in[9];
    const float* bwb  = (const float*)d_in[10];
    float* out = (float*)d_out;

    // Workspace layout (floats): pooled | ybuf | a_h | a_w  (~3.7 MB total)
    float* ws     = (float*)d_ws;
    float* pooled = ws;                                  // N*C*112
    float* ybuf   = pooled + (size_t)N_ * C_ * L_;       // N*8*112
    float* a_h    = ybuf + (size_t)N_ * MIP * L_;        // N*C*56
    float* a_w    = a_h + (size_t)N_ * C_ * H_;          // N*C*56

    pool_kernel<<<N_ * C_, 64, 0, stream>>>(x, pooled);
    conv1_kernel<<<N_ * 7, 32, 0, stream>>>(pooled, w1, b1, g1, be1, mu1, var1, ybuf);
    dim3 g3(16, 4, N_ * 2);
    attn_kernel<<<g3, 32, 0, stream>>>(ybuf, whw, bhb, www, bwb, a_h, a_w);
    const size_t total4 = (size_t)N_ * C_ * H_ * (W_ / 4);     // 3,211,264
    apply_kernel<<<(unsigned)(total4 / 256), 256, 0, stream>>>(x, a_h, a_w, out);
}